// NLinearIndividual_30966714204742
// MI455X (gfx1250) — compile-verified
//
#include <hip/hip_runtime.h>

// NLinear (per-channel linears), fp32, V_WMMA_F32_16X16X4_F32 + TDM staging.
//   out[b,f,c] = sum_l x[b,l,c]*W[c,f,l] + bias[c,f] + norm[b,c]*(1 - sum_l W[c,f,l])
// (algebraically identical to centering x by norm = x[:,L-1,:]).
// Per channel: GEMM M=128(batch) x N=192(forecast) x K=512(lookback).
// Block = 512 threads = 16 wave32s; wave w owns channel c0+w, one 16x16 tile.
// A tiles (x) are staged by the Tensor Data Mover into double-buffered LDS
// (pad_enable makes fragment reads bank-conflict-free); B rows stream from
// global as b64 fragments; W row-sums ride along in registers for the norm fix.

typedef __attribute__((ext_vector_type(2))) float v2f;
typedef __attribute__((ext_vector_type(8))) float v8f;
typedef __attribute__((ext_vector_type(4))) unsigned int u32x4;
typedef __attribute__((ext_vector_type(8))) int i32x8;
typedef __attribute__((ext_vector_type(4))) int i32x4;

constexpr int Bn = 128;   // batch
constexpr int Ln = 512;   // lookback (K)
constexpr int Cn = 512;   // channels
constexpr int Fn = 192;   // forecast

constexpr int CB     = 16;        // channels per block (= waves per block)
constexpr int KCHUNK = 32;        // K staged per TDM tile
// TDM tile: dim0=16(c) x dim1=32(l) x dim2=16(b) = 8192 dwords, +1 dword pad
// every 128 dwords -> 8256 dwords; padded index = raw + (raw>>7).
constexpr int TBUF   = 8256;
constexpr int RN     = 17;            // padded N stride for result tile
constexpr int RC     = 16 * RN + 4;   // 276: per-channel result stride

__device__ __forceinline__ unsigned lds_off(const void* p) {
  // generic LDS pointers carry the wave-relative LDS offset in the low 32 bits
  return (unsigned)(size_t)p;
}

__device__ __forceinline__ void tdm_load_tile(unsigned lds_addr,
                                              unsigned long long gaddr) {
  // D# group 0: count=1, lds_addr, 57-bit global byte addr, type=2
  u32x4 g0 = { 1u,
               lds_addr,
               (unsigned)gaddr,
               (unsigned)((gaddr >> 32) & 0x01FFFFFFu) | (2u << 30) };
  // D# group 1: data_size=4B(2), pad_enable, pad_interval=128dw(6), pad_amount=1dw(0)
  //   tensor_dim0=512(c), tensor_dim1=512(l), tile=(16,32,16),
  //   dim0_stride=512 (l step), dim1_stride=262144 (b step)
  i32x8 g1 = { (int)((2u << 16) | (1u << 20) | (6u << 22)),
               (int)(512u << 16),            // [31:16] tensor_dim0 lo
               (int)(512u << 16),            // [15:0] dim0 hi=0, [31:16] tensor_dim1 lo
               (int)(16u << 16),             // [15:0] dim1 hi=0, [31:16] tile_dim0
               (int)(32u | (16u << 16)),     // tile_dim1, tile_dim2
               512,                          // tensor_dim0_stride lo32
               0,                            // stride0 hi, stride1 lo16 (=0)
               4 };                          // stride1 bits 47:16 (262144>>16)
  // D# group 2: tensor_dim2=128(b), dim2_stride=262144
  i32x4 g2 = { 128, 0, 262144, 0 };
  i32x4 g3 = { 0, 0, 0, 0 };
  // this toolchain's builtin takes an extra (trailing) 8-dword group; the D#
  // group0 count=1 marks it unused -> zero-filled terminator.
  i32x8 g4 = { 0, 0, 0, 0, 0, 0, 0, 0 };
  __builtin_amdgcn_tensor_load_to_lds(g0, g1, g2, g3, g4, 0);
}

__global__ __launch_bounds__(512, 1)
void nlinear_wmma_f32(const float* __restrict__ x,
                      const float* __restrict__ W,
                      const float* __restrict__ bias,
                      float* __restrict__ out) {
  __shared__ __align__(128) float Abuf[2][TBUF];  // TDM double buffer
  __shared__ float Res[CB * RC];                  // [cc][m][n] result staging
  __shared__ float Norm[CB * 16];                 // x[b, L-1, c]

  const int t    = threadIdx.x;
  const int wave = t >> 5;           // channel within block
  const int lane = t & 31;
  const int ln   = lane & 15;        // N (B,C,D) / M (A) index
  const int hf   = lane >> 4;        // half-wave selector

  const int mtile = blockIdx.x & 7;
  const int ftile = blockIdx.x >> 3;
  const int c0    = blockIdx.y * CB;
  const int bbase = mtile * 16;
  const int fbase = ftile * 16;
  const int c     = c0 + wave;

  // tile origin in x: x[bbase, kc, c0]; kc advances by KCHUNK*Cn elements
  const unsigned long long gbase =
      (unsigned long long)(const void*)(x + (size_t)bbase * (Ln * Cn) + c0);

  // kick off the first TDM tile as early as possible (wave 0 only)
  if (wave == 0 && lane == 0) {
    tdm_load_tile(lds_off(&Abuf[0][0]), gbase);
  }

  // ---- stage norms: x[bbase+b, L-1, c0+cc], coalesced along cc ----
  if (t < CB * 16) {
    const int cc = t & 15, bb = t >> 4;
    Norm[cc * 16 + bb] =
        x[(size_t)(bbase + bb) * (Ln * Cn) + (size_t)(Ln - 1) * Cn + (c0 + cc)];
  }

  // B row for this lane: W[c, fbase+ln, :], contiguous in K
  const float* wrow = W + ((size_t)c * Fn + (fbase + ln)) * Ln;
  const float biasv = bias[(size_t)c * Fn + (fbase + ln)];

  v8f acc = {};
  float ssum = 0.0f;   // this lane's partial sum of its W row (half of K)
  int p = 0;

  for (int kc = 0; kc < Ln; kc += KCHUNK) {
    if (wave == 0 && lane == 0) {
      if (kc + KCHUNK < Ln) {
        tdm_load_tile(lds_off(&Abuf[p ^ 1][0]),
                      gbase + (unsigned long long)(kc + KCHUNK) * (Cn * 4));
        __builtin_amdgcn_s_wait_tensorcnt(1);   // current tile landed
      } else {
        __builtin_amdgcn_s_wait_tensorcnt(0);
      }
    }
    __syncthreads();   // current buffer visible to all waves

    // ---- batched fragment loads, then 8 x WMMA ----
    const float* lb = &Abuf[p][0];
    v2f bfr[8], afr[8];
    #pragma unroll
    for (int s = 0; s < 8; ++s) {
      bfr[s] = *(const v2f*)(wrow + kc + 4 * s + 2 * hf);
    }
    #pragma unroll
    for (int s = 0; s < 8; ++s) {
      const int k0   = 4 * s + 2 * hf;
      const int raw0 = ((ln << 5) + k0) * 16 + wave;   // [b][l][c] packed
      const int raw1 = raw0 + 16;                      // next l
      afr[s].x = lb[raw0 + (raw0 >> 7)];               // pad-adjusted index
      afr[s].y = lb[raw1 + (raw1 >> 7)];
    }
    #pragma unroll
    for (int s = 0; s < 8; ++s) {
      acc = __builtin_amdgcn_wmma_f32_16x16x4_f32(
          false, afr[s], false, bfr[s], (short)0, acc, false, false);
      ssum += bfr[s].x + bfr[s].y;
    }

    if (kc + KCHUNK < Ln) {
      __builtin_prefetch(wrow + kc + KCHUNK + 2 * hf, 0, 3);
    }
    __syncthreads();   // all waves done reading buf[p] -> reusable as DMA dest
    p ^= 1;
  }

  // full W row sum: combine the two half-wave partials
  const float S = ssum + __shfl_xor(ssum, 16, 32);
  const float cf = 1.0f - S;

  // ---- fold bias + norm correction, dump to LDS (lane=N, M = r + 8*hf) ----
  #pragma unroll
  for (int r = 0; r < 8; ++r) {
    const int m = r + 8 * hf;
    Res[wave * RC + m * RN + ln] = acc[r] + Norm[wave * 16 + m] * cf + biasv;
  }
  __syncthreads();

  // ---- cooperative coalesced store (c fastest) ----
  #pragma unroll
  for (int i = 0; i < (CB * 16 * 16) / 512; ++i) {
    const int idx = t + i * 512;
    const int cc  = idx & 15;
    const int nn  = (idx >> 4) & 15;
    const int mm  = idx >> 8;
    out[(size_t)(bbase + mm) * (Fn * Cn) + (size_t)(fbase + nn) * Cn + (c0 + cc)] =
        Res[cc * RC + mm * RN + nn];
  }
}

extern "C" void kernel_launch(void* const* d_in, const int* in_sizes, int n_in,
                              void* d_out, int out_size, void* d_ws, size_t ws_size,
                              hipStream_t stream) {
  const float* x    = (const float*)d_in[0];  // (128, 512, 512)
  const float* W    = (const float*)d_in[1];  // (512, 192, 512)
  const float* bias = (const float*)d_in[2];  // (512, 192)
  float* out        = (float*)d_out;          // (128, 192, 512)

  // x: fuse (mtile, ftile) so the 96 blocks sharing one 16-channel slice of
  // W and x are launch-adjacent (L2 reuse); y: channel blocks.
  dim3 grid((Bn / 16) * (Fn / 16), Cn / CB, 1);   // (96, 32)
  nlinear_wmma_f32<<<grid, 512, 0, stream>>>(x, W, bias, out);
}